// NSPhysicsLoss_39015482917700
// MI455X (gfx1250) — compile-verified
//
#include <hip/hip_runtime.h>

// ---------------- problem constants (from reference) ----------------
#define BATCH 32
#define H 512
#define W 512
#define TILE 32
#define HALO 2
#define LW (TILE + 2*HALO)       // 36 staged rows
#define SPX 40                   // LDS row stride (floats); 160B, keeps float4 alignment
#define CX 4                     // tile x_local=0 lives at LDS column 4 (halo at cols 2..3)
#define NTX (W / TILE)           // 16
#define NTY (H / TILE)           // 16
#define NBLK (NTX * NTY * BATCH) // 8192 tile-blocks
#define N_DATA  16777216.0f      // 32*2*512*512
#define N_PHYS  8388608.0f       // 32*512*512

__constant__ const float kNU = 0.001f;
__constant__ const float kInvDT = 100.0f;      // 1/DT, DT=0.01
__constant__ const float kLambda = 0.001f;

typedef __attribute__((ext_vector_type(2))) float v2f;
typedef __attribute__((ext_vector_type(8))) float v8f;

typedef __attribute__((address_space(1))) int* gptr_i32;   // global word ptr
typedef __attribute__((address_space(3))) int* lptr_i32;   // LDS word ptr

// --------- CDNA5 async global->LDS staging (probe-confirmed to exist) ----------
#if defined(__has_builtin)
#  if __has_builtin(__builtin_amdgcn_global_load_async_to_lds_b32) && \
      __has_builtin(__builtin_amdgcn_s_wait_asynccnt)
#    define USE_ASYNC_LDS 1
#  endif
#endif
#ifndef USE_ASYNC_LDS
#  define USE_ASYNC_LDS 0
#endif

__device__ __forceinline__ int clampi(int v, int lo, int hi) {
    return v < lo ? lo : (v > hi ? hi : v);
}

// ---- branchless jnp.gradient (one application), taps Wm=S[g-1], C=S[g], E=S[g+1]
__device__ __forceinline__ float sel_grad1(float Wm, float C, float E, int g, int N) {
    const float fwd = E - C, bwd = C - Wm;
    float r = 0.5f * (fwd + bwd);
    r = (g == 0)     ? fwd : r;
    r = (g == N - 1) ? bwd : r;
    return r;
}

// ---- branchless jnp.gradient applied twice along one axis
//      taps A=S[g-2], Bm=S[g-1], C=S[g], Dp=S[g+1], E=S[g+2]
__device__ __forceinline__ float sel_grad2(float A, float Bm, float C, float Dp,
                                           float E, int g, int N) {
    float r = 0.25f * ((E + A) - 2.0f * C);                // interior
    const float f0  = 0.5f * (E - C) - (Dp - C);           // g == 0
    const float f1  = 0.5f * (0.5f * (E - C) - (C - Bm));  // g == 1
    const float fN1 = (C - Bm) - 0.5f * (C - A);           // g == N-1
    const float fN2 = 0.5f * ((Dp - C) - 0.5f * (C - A));  // g == N-2
    r = (g == 0)     ? f0  : r;
    r = (g == 1)     ? f1  : r;
    r = (g == N - 1) ? fN1 : r;
    r = (g == N - 2) ? fN2 : r;
    return r;
}

// Wave-level exact f32 sum via V_WMMA_F32_16X16X4_F32 "ones-matmul":
// A vgpr0 = per-lane partial, A vgpr1 = 0 -> D[m][n] = PS[m] + PS[m+16] for all n.
// Summing the 8 D components leaves wave-total = (lane0 value) + (lane16 value).
__device__ __forceinline__ float wave_wmma_partial(float ps) {
    v2f a = {ps, 0.0f};
    v2f ones = {1.0f, 1.0f};
    v8f c = {0.0f, 0.0f, 0.0f, 0.0f, 0.0f, 0.0f, 0.0f, 0.0f};
    v8f d = __builtin_amdgcn_wmma_f32_16x16x4_f32(
        false, a, false, ones, (short)0, c, false, false);
    return ((d[0] + d[1]) + (d[2] + d[3])) + ((d[4] + d[5]) + (d[6] + d[7]));
}

// ---------------- main tile kernel ----------------
__global__ __launch_bounds__(256) void ns_loss_tiles(
    const float* __restrict__ u_pred,
    const float* __restrict__ u_true,
    const float* __restrict__ u_prev,
    float* __restrict__ partials)
{
    __shared__ __align__(16) float su[LW * SPX];
    __shared__ __align__(16) float sv[LW * SPX];
    __shared__ float sredD[16];
    __shared__ float sredP[16];

    const int tid = threadIdx.x;
    const int tx = blockIdx.x % NTX;
    const int ty = blockIdx.x / NTX;
    const int b  = blockIdx.y;
    const size_t HW = (size_t)H * W;
    const size_t base_u = (size_t)b * 2 * HW;
    const size_t base_v = base_u + HW;
    const int gy0 = ty * TILE - HALO;
    const int gx0 = tx * TILE - HALO;

    // ---- stage u_pred tile (+halo2, edge-clamped) into LDS via async copy ----
    for (int idx = tid; idx < LW * LW; idx += 256) {
        const int r = idx / LW, cc = idx % LW;           // cc: 0..35 -> x offset -2..33
        const int gy = clampi(gy0 + r, 0, H - 1);
        const int gx = clampi(gx0 + cc, 0, W - 1);
        const size_t go = (size_t)gy * W + gx;
        const int lo = r * SPX + cc + 2;                 // col = x_local + CX
#if USE_ASYNC_LDS
        __builtin_amdgcn_global_load_async_to_lds_b32(
            (gptr_i32)(u_pred + base_u + go), (lptr_i32)(&su[lo]), 0, 0);
        __builtin_amdgcn_global_load_async_to_lds_b32(
            (gptr_i32)(u_pred + base_v + go), (lptr_i32)(&sv[lo]), 0, 0);
#else
        su[lo] = u_pred[base_u + go];
        sv[lo] = u_pred[base_v + go];
#endif
    }
#if USE_ASYNC_LDS
    __builtin_amdgcn_s_wait_asynccnt(0);
#endif
    __syncthreads();

    // each thread owns a 1x4 pixel strip: row ly, columns lx4..lx4+3
    const int ly  = tid >> 3;
    const int lx4 = (tid & 7) << 2;
    const int r   = ly + HALO;
    const int c   = lx4 + CX;
    const int lo  = r * SPX + c;
    const int gy  = ty * TILE + ly;
    const int gx4 = tx * TILE + lx4;
    const size_t go = (size_t)gy * W + gx4;

    // pointwise streams: one float4 per tensor per channel (16B aligned)
    const float4 t0v = *(const float4*)(u_true + base_u + go);
    const float4 t1v = *(const float4*)(u_true + base_v + go);
    const float4 p0v = *(const float4*)(u_prev + base_u + go);
    const float4 p1v = *(const float4*)(u_prev + base_v + go);
    const float t0a[4] = {t0v.x, t0v.y, t0v.z, t0v.w};
    const float t1a[4] = {t1v.x, t1v.y, t1v.z, t1v.w};
    const float p0a[4] = {p0v.x, p0v.y, p0v.z, p0v.w};
    const float p1a[4] = {p1v.x, p1v.y, p1v.z, p1v.w};

    // ---- shared tap loads (aligned float4 LDS reads, valid for ALL tiles:
    //      halo cols/rows always exist; clamped cells only feed lanes whose
    //      selected boundary expression never uses them) ----
    const float4 uxm = *(const float4*)&su[lo - 4];
    const float4 uxc = *(const float4*)&su[lo];
    const float4 uxp = *(const float4*)&su[lo + 4];
    const float4 un1 = *(const float4*)&su[lo - SPX];
    const float4 us1 = *(const float4*)&su[lo + SPX];
    const float4 un2 = *(const float4*)&su[lo - 2*SPX];
    const float4 us2 = *(const float4*)&su[lo + 2*SPX];
    const float4 vxm = *(const float4*)&sv[lo - 4];
    const float4 vxc = *(const float4*)&sv[lo];
    const float4 vxp = *(const float4*)&sv[lo + 4];
    const float4 vn1 = *(const float4*)&sv[lo - SPX];
    const float4 vs1 = *(const float4*)&sv[lo + SPX];
    const float4 vn2 = *(const float4*)&sv[lo - 2*SPX];
    const float4 vs2 = *(const float4*)&sv[lo + 2*SPX];

    // a8[k] = field at x offset (k-2) relative to strip start
    const float au[8] = {uxm.z, uxm.w, uxc.x, uxc.y, uxc.z, uxc.w, uxp.x, uxp.y};
    const float av[8] = {vxm.z, vxm.w, vxc.x, vxc.y, vxc.z, vxc.w, vxp.x, vxp.y};
    const float un1a[4] = {un1.x, un1.y, un1.z, un1.w};
    const float us1a[4] = {us1.x, us1.y, us1.z, us1.w};
    const float un2a[4] = {un2.x, un2.y, un2.z, un2.w};
    const float us2a[4] = {us2.x, us2.y, us2.z, us2.w};
    const float vn1a[4] = {vn1.x, vn1.y, vn1.z, vn1.w};
    const float vs1a[4] = {vs1.x, vs1.y, vs1.z, vs1.w};
    const float vn2a[4] = {vn2.x, vn2.y, vn2.z, vn2.w};
    const float vs2a[4] = {vs2.x, vs2.y, vs2.z, vs2.w};

    float accD = 0.0f, accP = 0.0f;

    const bool interior = (tx > 0) & (tx < NTX - 1) & (ty > 0) & (ty < NTY - 1);
    if (interior) {
        // ---- minimal branch-free path (196 of 256 tile positions) ----
#pragma unroll
        for (int j = 0; j < 4; ++j) {
            const float u = au[j + 2];
            const float v = av[j + 2];

            const float d0 = u - t0a[j], d1 = v - t1a[j];
            accD += d0 * d0 + d1 * d1;

            const float ut = (u - p0a[j]) * kInvDT;
            const float vt = (v - p1a[j]) * kInvDT;

            const float ux = 0.5f * (au[j + 3] - au[j + 1]);
            const float vx = 0.5f * (av[j + 3] - av[j + 1]);
            const float uy = 0.5f * (us1a[j] - un1a[j]);
            const float vy = 0.5f * (vs1a[j] - vn1a[j]);

            const float lap_u = 0.25f * ((au[j + 4] + au[j]) + (us2a[j] + un2a[j]) - 4.0f * u);
            const float lap_v = 0.25f * ((av[j + 4] + av[j]) + (vs2a[j] + vn2a[j]) - 4.0f * v);

            const float ru = ut + u * ux + v * uy - kNU * lap_u;
            const float rv = vt + u * vx + v * vy - kNU * lap_v;
            accP += ru * ru + rv * rv;
        }
    } else {
        // ---- edge tiles: same tap registers, branchless select of the exact
        //      jnp.gradient boundary expressions (v_cndmask, no branches) ----
#pragma unroll
        for (int j = 0; j < 4; ++j) {
            const int gx = gx4 + j;
            const float u = au[j + 2];
            const float v = av[j + 2];

            const float d0 = u - t0a[j], d1 = v - t1a[j];
            accD += d0 * d0 + d1 * d1;

            const float ut = (u - p0a[j]) * kInvDT;
            const float vt = (v - p1a[j]) * kInvDT;

            const float ux = sel_grad1(au[j + 1], u, au[j + 3], gx, W);
            const float vx = sel_grad1(av[j + 1], v, av[j + 3], gx, W);
            const float uy = sel_grad1(un1a[j], u, us1a[j], gy, H);
            const float vy = sel_grad1(vn1a[j], v, vs1a[j], gy, H);

            const float lap_u = sel_grad2(au[j], au[j + 1], u, au[j + 3], au[j + 4], gx, W)
                              + sel_grad2(un2a[j], un1a[j], u, us1a[j], us2a[j], gy, H);
            const float lap_v = sel_grad2(av[j], av[j + 1], v, av[j + 3], av[j + 4], gx, W)
                              + sel_grad2(vn2a[j], vn1a[j], v, vs1a[j], vs2a[j], gy, H);

            const float ru = ut + u * ux + v * uy - kNU * lap_u;
            const float rv = vt + u * vx + v * vy - kNU * lap_v;
            accP += ru * ru + rv * rv;
        }
    }

    // ---- in-wave exact reduction on the matrix pipe (EXEC all-1s; the
    //      interior/boundary branch above is block-uniform) ----
    const float sD = wave_wmma_partial(accD);
    const float sP = wave_wmma_partial(accP);

    if ((tid & 15) == 0) {         // lanes 0 & 16 of each of the 8 waves
        sredD[tid >> 4] = sD;
        sredP[tid >> 4] = sP;
    }
    __syncthreads();

    if (tid == 0) {
        float td = 0.0f, tp = 0.0f;
#pragma unroll
        for (int k = 0; k < 16; ++k) { td += sredD[k]; tp += sredP[k]; }
        const int blk = blockIdx.y * gridDim.x + blockIdx.x;
        partials[2 * blk + 0] = td;
        partials[2 * blk + 1] = tp;
    }
}

// ---------------- deterministic finalize ----------------
__global__ __launch_bounds__(256) void ns_loss_finalize(
    const float* __restrict__ partials, float* __restrict__ out)
{
    __shared__ float sd[256];
    __shared__ float sp[256];
    const int tid = threadIdx.x;
    float d = 0.0f, p = 0.0f;
    for (int i = tid; i < NBLK; i += 256) {   // fixed order -> deterministic
        d += partials[2 * i + 0];
        p += partials[2 * i + 1];
    }
    sd[tid] = d; sp[tid] = p;
    __syncthreads();
    for (int s = 128; s > 0; s >>= 1) {
        if (tid < s) { sd[tid] += sd[tid + s]; sp[tid] += sp[tid + s]; }
        __syncthreads();
    }
    if (tid == 0) {
        out[0] = sd[0] * (1.0f / N_DATA) + kLambda * (sp[0] * (1.0f / N_PHYS));
    }
}

extern "C" void kernel_launch(void* const* d_in, const int* in_sizes, int n_in,
                              void* d_out, int out_size, void* d_ws, size_t ws_size,
                              hipStream_t stream) {
    const float* u_pred = (const float*)d_in[0];
    const float* u_true = (const float*)d_in[1];
    const float* u_prev = (const float*)d_in[2];
    float* out = (float*)d_out;
    float* partials = (float*)d_ws;   // 2 * NBLK floats = 64 KB

    dim3 grid(NTX * NTY, BATCH);
    ns_loss_tiles<<<grid, 256, 0, stream>>>(u_pred, u_true, u_prev, partials);
    ns_loss_finalize<<<1, 256, 0, stream>>>(partials, out);
}